// GNN_79482664780161
// MI455X (gfx1250) — compile-verified
//
#include <hip/hip_runtime.h>
#include <hip/hip_bf16.h>

#define D 128
#define BN_EPS 1e-5f

typedef __attribute__((ext_vector_type(2))) float v2f;
typedef __attribute__((ext_vector_type(8))) float v8f;

// ---------------------------------------------------------------------------
// h[i] = x_emb1[x[i,0]] + x_emb2[x[i,1]]
// ---------------------------------------------------------------------------
__global__ void node_embed_kernel(const int* __restrict__ x,
                                  const float* __restrict__ xe1,
                                  const float* __restrict__ xe2,
                                  float* __restrict__ h, int N) {
    int i = blockIdx.x * blockDim.x + threadIdx.x;
    if (i >= N * D) return;
    int node = i >> 7;
    int f = i & (D - 1);
    int a = x[2 * node];
    int b = x[2 * node + 1];
    h[i] = xe1[a * D + f] + xe2[b * D + f];
}

// ---------------------------------------------------------------------------
// Self-loop init: agg[i] = h[i] + ee1[bond=4] + ee2[dir=0]
// ---------------------------------------------------------------------------
__global__ void agg_init_kernel(const float* __restrict__ h,
                                const float* __restrict__ ee1_self,
                                const float* __restrict__ ee2_self,
                                float* __restrict__ agg, int N) {
    int i = blockIdx.x * blockDim.x + threadIdx.x;
    if (i >= N * D) return;
    int f = i & (D - 1);
    agg[i] = h[i] + ee1_self[f] + ee2_self[f];
}

// ---------------------------------------------------------------------------
// Per-edge message + scatter-add. One block = one edge, thread = feature.
// msg = h[col] + ee1[type] + ee2[dir]; atomicAdd into agg[row].
// ---------------------------------------------------------------------------
__global__ void edge_message_kernel(const float* __restrict__ h,
                                    const int* __restrict__ edge_index, // [2,E]
                                    const int* __restrict__ edge_attr,  // [E,2]
                                    const float* __restrict__ ee1,      // [6,D]
                                    const float* __restrict__ ee2,      // [3,D]
                                    float* __restrict__ agg, int E) {
    int e = blockIdx.x;
    if (e >= E) return;
    int f = threadIdx.x;
    int r = edge_index[e];          // row (aggregate target)
    int c = edge_index[E + e];      // col (gather source)
    int t1 = edge_attr[2 * e];
    int t2 = edge_attr[2 * e + 1];
    float msg = h[(size_t)c * D + f] + ee1[t1 * D + f] + ee2[t2 * D + f];
    atomicAdd(&agg[(size_t)r * D + f], msg);
}

// ---------------------------------------------------------------------------
// Fused GIN MLP on a 16-node tile using V_WMMA_F32_16X16X4_F32.
//   hidden = relu(agg @ W1 + b1)   [16 x 256], kept in LDS
//   out    = hidden @ W2 + b2      [16 x 128]
// Block = 128 threads = 4 wave32s. EXEC is all-ones for every WMMA.
// A-frag (16x4 f32): lane L, vgpr g -> A[L%16][k + 2*(L/16) + g]
// B-frag (4x16 f32): lane L, vgpr g -> B[k + 2*(L/16) + g][L%16]
// C/D (16x16 f32) : elem g -> row g + 8*(L/16), col L%16
// ---------------------------------------------------------------------------
__global__ void __launch_bounds__(128)
gin_mlp_kernel(const float* __restrict__ agg,
               const float* __restrict__ W1, const float* __restrict__ b1,
               const float* __restrict__ W2, const float* __restrict__ b2,
               float* __restrict__ out) {
    // pad strides so row stride % 64 banks == 4 -> conflict-free column reads
    __shared__ float As[16][D + 4];
    __shared__ float Hs[16][2 * D + 4];

    const int tid  = threadIdx.x;
    const int wave = tid >> 5;
    const int lane = tid & 31;
    const int half = lane >> 4;   // 0: lanes 0-15, 1: lanes 16-31
    const int l16  = lane & 15;
    const int node0 = blockIdx.x * 16;

    // Stage A tile (16 x 128) into LDS, coalesced.
    for (int i = tid; i < 16 * D; i += 128) {
        As[i >> 7][i & (D - 1)] = agg[(size_t)node0 * D + i];
    }
    __syncthreads();

    // ---- GEMM1: [16x128] @ [128x256] -> Hs, bias + ReLU ----
    for (int t = 0; t < 4; ++t) {
        const int n0 = (wave * 4 + t) * 16;
        v8f acc = {};
        for (int k = 0; k < D; k += 4) {
            const int ka = k + 2 * half;
            v2f a, b;
            a.x = As[l16][ka];
            a.y = As[l16][ka + 1];
            b.x = W1[(size_t)ka * (2 * D) + n0 + l16];
            b.y = W1[(size_t)(ka + 1) * (2 * D) + n0 + l16];
            acc = __builtin_amdgcn_wmma_f32_16x16x4_f32(
                false, a, false, b, (short)0, acc, false, false);
        }
        const float bias = b1[n0 + l16];
        #pragma unroll
        for (int g = 0; g < 8; ++g) {
            float v = acc[g] + bias;
            Hs[g + 8 * half][n0 + l16] = v > 0.0f ? v : 0.0f;
        }
    }
    __syncthreads();

    // ---- GEMM2: [16x256] @ [256x128] -> out, bias ----
    for (int t = 0; t < 2; ++t) {
        const int n0 = (wave * 2 + t) * 16;
        v8f acc = {};
        for (int k = 0; k < 2 * D; k += 4) {
            const int ka = k + 2 * half;
            v2f a, b;
            a.x = Hs[l16][ka];
            a.y = Hs[l16][ka + 1];
            b.x = W2[(size_t)ka * D + n0 + l16];
            b.y = W2[(size_t)(ka + 1) * D + n0 + l16];
            acc = __builtin_amdgcn_wmma_f32_16x16x4_f32(
                false, a, false, b, (short)0, acc, false, false);
        }
        const float bias = b2[n0 + l16];
        #pragma unroll
        for (int g = 0; g < 8; ++g) {
            const int m = g + 8 * half;
            out[(size_t)(node0 + m) * D + n0 + l16] = acc[g] + bias;
        }
    }
}

// ---------------------------------------------------------------------------
// BatchNorm statistics: per-feature sum and sum-of-squares.
// ---------------------------------------------------------------------------
__global__ void bn_zero_kernel(float* __restrict__ stats) {
    stats[threadIdx.x] = 0.0f;   // 256 threads: sum[128], sumsq[128]
}

__global__ void bn_stats_kernel(const float* __restrict__ h,
                                float* __restrict__ stats, int N) {
    const int f = threadIdx.x;   // 128 threads = features
    float s = 0.0f, s2 = 0.0f;
    for (int r = blockIdx.x; r < N; r += gridDim.x) {
        float v = h[(size_t)r * D + f];
        s += v;
        s2 += v * v;
    }
    atomicAdd(&stats[f], s);
    atomicAdd(&stats[D + f], s2);
}

__global__ void bn_apply_kernel(float* __restrict__ h,
                                const float* __restrict__ stats,
                                const float* __restrict__ gamma,
                                const float* __restrict__ beta,
                                int N, int do_relu) {
    int i = blockIdx.x * blockDim.x + threadIdx.x;
    if (i >= N * D) return;
    const int f = i & (D - 1);
    const float invN = 1.0f / (float)N;
    const float mean = stats[f] * invN;
    const float var = stats[D + f] * invN - mean * mean;
    float v = gamma[f] * (h[i] - mean) * rsqrtf(var + BN_EPS) + beta[f];
    if (do_relu) v = fmaxf(v, 0.0f);
    h[i] = v;
}

// ---------------------------------------------------------------------------
extern "C" void kernel_launch(void* const* d_in, const int* in_sizes, int n_in,
                              void* d_out, int out_size, void* d_ws, size_t ws_size,
                              hipStream_t stream) {
    const int*   x     = (const int*)d_in[0];
    const int*   ei    = (const int*)d_in[1];
    const int*   ea    = (const int*)d_in[2];
    const float* xe1   = (const float*)d_in[3];
    const float* xe2   = (const float*)d_in[4];
    const float* ee1   = (const float*)d_in[5];   // [5,6,D]
    const float* ee2   = (const float*)d_in[6];   // [5,3,D]
    const float* W1    = (const float*)d_in[7];   // [5,D,2D]
    const float* b1    = (const float*)d_in[8];   // [5,2D]
    const float* W2    = (const float*)d_in[9];   // [5,2D,D]
    const float* b2    = (const float*)d_in[10];  // [5,D]
    const float* gamma = (const float*)d_in[11];  // [5,D]
    const float* beta  = (const float*)d_in[12];  // [5,D]
    float* out = (float*)d_out;

    const int N = in_sizes[0] / 2;
    const int E = in_sizes[1] / 2;

    float* h    = (float*)d_ws;                 // [N, D]
    float* agg  = h + (size_t)N * D;            // [N, D]
    float* stats = agg + (size_t)N * D;         // [2*D]

    const int elems = N * D;
    const int eb = (elems + 255) / 256;

    node_embed_kernel<<<eb, 256, 0, stream>>>(x, xe1, xe2, h, N);

    for (int l = 0; l < 5; ++l) {
        const float* ee1_l = ee1 + (size_t)l * 6 * D;
        const float* ee2_l = ee2 + (size_t)l * 3 * D;

        // self loops: bond type 4, direction 0
        agg_init_kernel<<<eb, 256, 0, stream>>>(h, ee1_l + 4 * D, ee2_l, agg, N);
        edge_message_kernel<<<E, D, 0, stream>>>(h, ei, ea, ee1_l, ee2_l, agg, E);

        float* mlp_out = (l == 4) ? out : h;
        gin_mlp_kernel<<<N / 16, 128, 0, stream>>>(
            agg,
            W1 + (size_t)l * D * 2 * D, b1 + (size_t)l * 2 * D,
            W2 + (size_t)l * 2 * D * D, b2 + (size_t)l * D,
            mlp_out);

        bn_zero_kernel<<<1, 2 * D, 0, stream>>>(stats);
        bn_stats_kernel<<<512, D, 0, stream>>>(mlp_out, stats, N);
        bn_apply_kernel<<<eb, 256, 0, stream>>>(mlp_out, stats,
                                                gamma + (size_t)l * D,
                                                beta + (size_t)l * D,
                                                N, (l < 4) ? 1 : 0);
    }
}